// GlobalAverageBlock_5669356831305
// MI455X (gfx1250) — compile-verified
//
#include <hip/hip_runtime.h>

typedef __attribute__((ext_vector_type(2))) float v2f;
typedef __attribute__((ext_vector_type(8))) float v8f;

#define WAVES_PER_BLOCK 8
#define BLOCK_THREADS 256

// ---------------------------------------------------------------------------
// Kernel 1: exclusive prefix sum of batch_lengths -> segment start offsets.
// One block, Hillis-Steele scan in LDS. B <= 1024.
// ---------------------------------------------------------------------------
__global__ void seg_offsets_kernel(const int* __restrict__ lens,
                                   int* __restrict__ offs, int B) {
  __shared__ int buf[1024];
  const int t = threadIdx.x;
  const int v = (t < B) ? lens[t] : 0;
  buf[t] = v;
  __syncthreads();
#pragma unroll
  for (int s = 1; s < 1024; s <<= 1) {
    const int add = (t >= s) ? buf[t - s] : 0;
    __syncthreads();
    buf[t] += add;
    __syncthreads();
  }
  if (t < B) offs[t] = buf[t] - v;  // exclusive scan
}

// ---------------------------------------------------------------------------
// Kernel 2 (specialized): segment mean via WMMA column-sum, D compile-time.
//   D_acc = ones(16x4) x X(4x16) + C  accumulates 4-row column sums.
// One workgroup per segment; wave w owns 16-column tile w.
// Addressing: uniform SGPR base (x) + 32-bit signed element offset so the
// compiler can use the GVS mode (saddr + voffset + IOFFSET immediates) with
// a single 32-bit induction-variable add per unrolled body.
// ---------------------------------------------------------------------------
template <int D>
__global__ void __launch_bounds__(BLOCK_THREADS)
seg_mean_wmma_kernel_t(const float* __restrict__ x,
                       const int* __restrict__ lens,
                       const int* __restrict__ offs,
                       float* __restrict__ out) {
  static_assert(D == 16 * WAVES_PER_BLOCK, "one 16-col tile per wave");

  const int seg  = blockIdx.x;
  const int len  = lens[seg];
  const int start = offs[seg];

  const int wave = threadIdx.x >> 5;   // wave32
  const int lane = threadIdx.x & 31;
  const int half = lane >> 4;          // 0: rows {+0,+1}, 1: rows {+2,+3}
  const int col  = lane & 15;
  const int c0   = wave * 16;

  // All-ones 16x4 A matrix: 2 f32 per lane, value 1.0 everywhere.
  const v2f a = {1.0f, 1.0f};

  // 32-bit element offset of (row = start + 2*half, col = c0 + col).
  // Max N*D = 2^27 elements, so this and all +k*D offsets fit in int32.
  int off = (start + 2 * half) * D + c0 + col;

  v8f acc0 = {};
  v8f acc1 = {};

  int r = 0;
  // Main loop: 8 rows per iteration, two independent WMMA chains.
  // Row strides are compile-time immediates (0, D, 4D, 5D elements).
  for (; r + 8 <= len; r += 8) {
    __builtin_prefetch(x + (off + 16 * D), 0, 1);
    v2f b0, b1;
    b0.x = x[off];
    b0.y = x[off + D];
    b1.x = x[off + 4 * D];
    b1.y = x[off + 5 * D];
    acc0 = __builtin_amdgcn_wmma_f32_16x16x4_f32(
        false, a, false, b0, (short)0, acc0, false, false);
    acc1 = __builtin_amdgcn_wmma_f32_16x16x4_f32(
        false, a, false, b1, (short)0, acc1, false, false);
    off += 8 * D;
  }
  // Masked tail: 4 rows per step (at most 2 steps).
  for (; r < len; r += 4) {
    const int row0 = r + 2 * half;
    const int row1 = row0 + 1;
    v2f b;
    b.x = (row0 < len) ? x[off] : 0.0f;
    b.y = (row1 < len) ? x[off + D] : 0.0f;
    acc0 = __builtin_amdgcn_wmma_f32_16x16x4_f32(
        false, a, false, b, (short)0, acc0, false, false);
    off += 4 * D;
  }

  // Every output row of the WMMA D-matrix holds the identical column sum;
  // element 0 of the accumulator on lane L is column (L & 15).
  const float sum = acc0[0] + acc1[0];
  if (lane < 16) {
    out[seg * D + c0 + lane] = sum / (float)len;
  }
}

// ---------------------------------------------------------------------------
// Kernel 2 (generic): runtime D fallback (multiple of 16).
// ---------------------------------------------------------------------------
__global__ void __launch_bounds__(BLOCK_THREADS)
seg_mean_wmma_kernel_g(const float* __restrict__ x,
                       const int* __restrict__ lens,
                       const int* __restrict__ offs,
                       float* __restrict__ out, int D) {
  const int seg  = blockIdx.x;
  const int len  = lens[seg];
  const long long start = (long long)offs[seg];

  const int wave = threadIdx.x >> 5;
  const int lane = threadIdx.x & 31;
  const int half = lane >> 4;
  const int col  = lane & 15;

  const v2f a = {1.0f, 1.0f};

  for (int tile = wave; tile * 16 < D; tile += WAVES_PER_BLOCK) {
    const int c0 = tile * 16;
    const float* __restrict__ q =
        x + (start + (long long)(2 * half)) * (long long)D + c0 + col;

    v8f acc0 = {};
    v8f acc1 = {};

    int r = 0;
    for (; r + 8 <= len; r += 8) {
      __builtin_prefetch(q + (long long)16 * D, 0, 1);
      v2f b0, b1;
      b0.x = q[0];
      b0.y = q[D];
      b1.x = q[(long long)4 * D];
      b1.y = q[(long long)5 * D];
      acc0 = __builtin_amdgcn_wmma_f32_16x16x4_f32(
          false, a, false, b0, (short)0, acc0, false, false);
      acc1 = __builtin_amdgcn_wmma_f32_16x16x4_f32(
          false, a, false, b1, (short)0, acc1, false, false);
      q += (long long)8 * D;
    }
    for (; r < len; r += 4) {
      const int row0 = r + 2 * half;
      const int row1 = row0 + 1;
      v2f b;
      b.x = (row0 < len) ? q[0] : 0.0f;
      b.y = (row1 < len) ? q[D] : 0.0f;
      acc0 = __builtin_amdgcn_wmma_f32_16x16x4_f32(
          false, a, false, b, (short)0, acc0, false, false);
      q += (long long)4 * D;
    }

    const float sum = acc0[0] + acc1[0];
    if (lane < 16) {
      out[(long long)seg * D + c0 + lane] = sum / (float)len;
    }
  }
}

// ---------------------------------------------------------------------------
// Host launcher
// ---------------------------------------------------------------------------
extern "C" void kernel_launch(void* const* d_in, const int* in_sizes, int n_in,
                              void* d_out, int out_size, void* d_ws, size_t ws_size,
                              hipStream_t stream) {
  const float* x    = (const float*)d_in[0];
  const int*   lens = (const int*)d_in[1];
  float*       out  = (float*)d_out;

  const int B = in_sizes[1];        // number of segments
  const int D = out_size / B;       // feature dim (128 in this problem)

  int* offs = (int*)d_ws;           // B ints of scratch

  seg_offsets_kernel<<<1, 1024, 0, stream>>>(lens, offs, B);

  if (D == 128) {
    seg_mean_wmma_kernel_t<128><<<B, BLOCK_THREADS, 0, stream>>>(
        x, lens, offs, out);
  } else {
    seg_mean_wmma_kernel_g<<<B, BLOCK_THREADS, 0, stream>>>(
        x, lens, offs, out, D);
  }
}